// CustomConv1D_dtoq_17411797418711
// MI455X (gfx1250) — compile-verified
//
#include <hip/hip_runtime.h>
#include <hip/hip_bf16.h>

typedef __attribute__((ext_vector_type(2))) float v2f;
typedef __attribute__((ext_vector_type(4))) float v4f;
typedef __attribute__((ext_vector_type(8))) float v8f;

#define KW       64
#define NC       10                      // channels
#define FLATK    (KW * NC)               // 640 flat reduction per window
#define TILE_M   16                      // windows per wave == WMMA M
#define CHUNK    64                      // flat-K floats staged per iteration
#define NCHUNK   (FLATK / CHUNK)         // 10
#define ROWSTRIDE 68                     // dwords per window row in LDS (pad: 16B-aligned, bank-conflict-free)
#define WAVES_PER_BLOCK 8
#define TILE_LDS_DW (TILE_M * ROWSTRIDE) // 1088
#define WAVE_LDS_DW (TILE_LDS_DW + FLATK) // 1728 dwords (tile + private weight copy)

// ---------- kernel 1: nealmon softmax weights, expanded to wflat[640] ----------
__global__ void nealmon_weights_kernel(const float* __restrict__ p5p,
                                       const float* __restrict__ p6p,
                                       float* __restrict__ wflat) {
  __shared__ float sh[KW];
  const int t = threadIdx.x;             // 0..63
  const float p5 = p5p[0], p6 = p6p[0];
  const float i  = (float)(t + 1);
  const float ll = p5 * i + p6 * i * i;
  sh[t] = ll;
  __syncthreads();
  float mx = sh[0];
  for (int j = 1; j < KW; ++j) mx = fmaxf(mx, sh[j]);   // redundant per-thread, trivial size
  __syncthreads();
  const float e = __expf(ll - mx);
  sh[t] = e;
  __syncthreads();
  float s = 0.f;
  for (int j = 0; j < KW; ++j) s += sh[j];
  const float w = e / s;
  for (int c = 0; c < NC; ++c) wflat[t * NC + c] = w;   // wflat[f] = w[f/10]
}

// ---------- kernel 2: streaming weighted reduction via v_wmma_f32_16x16x4_f32 ----------
__global__ void __launch_bounds__(256)
conv1d_wmma_kernel(const float* __restrict__ x, const float* __restrict__ wflat,
                   float* __restrict__ out, int ntiles) {
  __shared__ float smem[WAVE_LDS_DW * WAVES_PER_BLOCK];  // 55296 B
  const int lane = threadIdx.x & 31;
  const int wave = threadIdx.x >> 5;
  const int tile = blockIdx.x * WAVES_PER_BLOCK + wave;
  if (tile >= ntiles) return;            // wave-uniform branch; EXEC stays all-1s below

  float* tileLds = smem + wave * WAVE_LDS_DW;
  float* wLds    = tileLds + TILE_LDS_DW;

  // Preload this wave's private copy of the expanded weights (640 f32 = 160 float4).
  for (int j = 0; j < 5; ++j) {
    const int f = j * 32 + lane;                         // float4 index 0..159
    v4f v = *(const v4f*)(wflat + (long)f * 4);
    *(v4f*)(wLds + f * 4) = v;
  }
  __builtin_amdgcn_wave_barrier();       // compiler fence; same-wave DS is HW in-order

  const long g0 = (long)tile * TILE_M;   // first window of this tile
  const float* xbase = x + g0 * (long)FLATK;  // windows are contiguous 640-float segments

  const int m = lane & 15;               // A-operand row (window within tile)
  const int h = lane >> 4;               // K-half: lanes 0-15 -> K=0,1 ; 16-31 -> K=2,3

  v8f acc = {};                          // C/D accumulator, 16x16 f32

  for (int chunk = 0; chunk < NCHUNK; ++chunk) {
    const float* gsrc = xbase + chunk * CHUNK;

    // Prefetch next chunk (one cacheline per window) -> global_prefetch_b8
    if (chunk + 1 < NCHUNK) {
      const float* nxt = xbase + (chunk + 1) * CHUNK + (long)(lane & 15) * FLATK;
      __builtin_prefetch(nxt, 0, 0);
    }

    // Coalesced global -> LDS stage: 16 windows x 64 floats = 256 float4, 8 per lane.
#pragma unroll
    for (int j = 0; j < 8; ++j) {
      const int f = j * 32 + lane;       // 0..255
      const int w = f >> 4;              // window 0..15
      const int q = f & 15;              // float4 within the 64-float chunk
      v4f v = *(const v4f*)(gsrc + (long)w * FLATK + q * 4);
      *(v4f*)(tileLds + w * ROWSTRIDE + q * 4) = v;    // 16B aligned (ROWSTRIDE*4 = 272)
    }
    __builtin_amdgcn_wave_barrier();

    // 16 WMMA steps of K=4 consume the 64-float chunk.
    // A (16x4 f32): lane m, VGPR pair = {A[m, 4kc+2h], A[m, 4kc+2h+1]}
    // B (4x16 f32): all columns identical = wflat, so D columns all equal the dot product.
#pragma unroll
    for (int kc = 0; kc < 16; ++kc) {
      v2f a = *(const v2f*)(tileLds + m * ROWSTRIDE + kc * 4 + h * 2);   // conflict-free banks
      v2f b = *(const v2f*)(wLds + chunk * CHUNK + kc * 4 + h * 2);      // LDS broadcast read
      acc = __builtin_amdgcn_wmma_f32_16x16x4_f32(
          /*neg_a=*/false, a, /*neg_b=*/false, b,
          /*c_mod=*/(short)0, acc, /*reuse_a=*/false, /*reuse_b=*/false);
    }
    __builtin_amdgcn_wave_barrier();
  }

  // D layout: VGPR v, lanes 0-15 hold M=v, lanes 16-31 hold M=v+8; value uniform over N.
  // Lane with (lane&15)==c < 10 writes channel c of its half's 8 windows.
  const int c = lane & 15;
  if (c < NC) {
#pragma unroll
    for (int v = 0; v < 8; ++v) {
      const long g = g0 + v + 8 * h;     // global window index
      out[g * NC + c] = acc[v];
    }
  }
}

extern "C" void kernel_launch(void* const* d_in, const int* in_sizes, int n_in,
                              void* d_out, int out_size, void* d_ws, size_t ws_size,
                              hipStream_t stream) {
  const float* x  = (const float*)d_in[0];
  const float* p5 = (const float*)d_in[1];
  const float* p6 = (const float*)d_in[2];
  float* wflat = (float*)d_ws;           // 640 f32 scratch
  float* out   = (float*)d_out;

  // 1) expand softmax weights into d_ws
  nealmon_weights_kernel<<<1, KW, 0, stream>>>(p5, p6, wflat);

  // 2) streaming WMMA reduction
  const long nwindows = (long)in_sizes[0] / FLATK;   // 131072 for the reference shapes
  const int  ntiles   = (int)(nwindows / TILE_M);    // 8192
  const int  nblocks  = (ntiles + WAVES_PER_BLOCK - 1) / WAVES_PER_BLOCK;  // 1024
  conv1d_wmma_kernel<<<nblocks, 256, 0, stream>>>(x, wflat, out, ntiles);
}